// ReliabilityDiagram_44813688766553
// MI455X (gfx1250) — compile-verified
//
#include <hip/hip_runtime.h>

#define NB 10
#define BLK 256
#define MAXBLOCKS 2048

// Native clang vector types: required by __builtin_nontemporal_load and
// guaranteed to lower to single b128 loads.
typedef float v4f __attribute__((ext_vector_type(4)));
typedef int v4i __attribute__((ext_vector_type(4)));

// 160-byte accumulator block at the start of d_ws.
struct RdWs {
  unsigned long long conf_fix[NB];  // sum of sigmoid(conf) in 2^-24 fixed point
  unsigned int cnt[NB];             // element count per bin
  unsigned int pos[NB];             // positive-label count per bin
};

__global__ void rd_init(RdWs* __restrict__ ws) {
  unsigned* w = (unsigned*)ws;
  int t = threadIdx.x;
  if (t < (int)(sizeof(RdWs) / 4)) w[t] = 0u;
}

__device__ __forceinline__ void rd_proc(float x, int lab, int tid,
                                        float* __restrict__ s_conf,
                                        unsigned* __restrict__ s_cp) {
  // sigmoid via v_exp_f32 + v_rcp_f32 (transcendental co-exec path on CDNA5)
  float e = __expf(-x);
  float conf = __builtin_amdgcn_rcpf(1.0f + e);
  // bins are (l,u] over linspace(0,1,11): idx = ceil(conf*10)-1, clipped
  int idx = (int)__builtin_ceilf(conf * 10.0f) - 1;
  idx = idx < 0 ? 0 : (idx > NB - 1 ? NB - 1 : idx);
  // Per-thread private LDS slot: bank = (idx*256 + tid) % 64 = tid % 64
  // -> all 32 lanes hit distinct banks, no conflicts, no atomics.
  int off = idx * BLK + tid;
  s_conf[off] += conf;
  s_cp[off] += 0x10000u + (unsigned)lab;  // count in hi16, positives in lo16
}

__global__ __launch_bounds__(BLK) void rd_accum(const float* __restrict__ logits,
                                                const int* __restrict__ labels,
                                                RdWs* __restrict__ ws, int n) {
  __shared__ float s_conf[NB * BLK];     // 10 KB
  __shared__ unsigned s_cp[NB * BLK];    // 10 KB
  const int tid = threadIdx.x;

#pragma unroll
  for (int b = 0; b < NB; ++b) {
    s_conf[b * BLK + tid] = 0.0f;
    s_cp[b * BLK + tid] = 0u;
  }
  __syncthreads();

  const int n4 = n >> 2;
  const int stride = (int)gridDim.x * BLK;
  const v4f* lg4 = (const v4f*)logits;
  const v4i* lb4 = (const v4i*)labels;

  // Streaming loop: two b128 non-temporal loads per iteration (data is
  // touched exactly once; NT keeps the 256MB stream from thrashing L2).
  for (int i = (int)blockIdx.x * BLK + tid; i < n4; i += stride) {
    v4f x = __builtin_nontemporal_load(&lg4[i]);
    v4i l = __builtin_nontemporal_load(&lb4[i]);
    rd_proc(x.x, l.x, tid, s_conf, s_cp);
    rd_proc(x.y, l.y, tid, s_conf, s_cp);
    rd_proc(x.z, l.z, tid, s_conf, s_cp);
    rd_proc(x.w, l.w, tid, s_conf, s_cp);
  }

  // Scalar remainder (n % 4), handled by the first `rem` global threads.
  int rem = n - (n4 << 2);
  int gid = (int)blockIdx.x * BLK + tid;
  if (gid < rem) {
    rd_proc(logits[(n4 << 2) + gid], labels[(n4 << 2) + gid], tid, s_conf, s_cp);
  }
  __syncthreads();

  // Tree-reduce the 256 per-thread histograms (fixed order -> deterministic).
  for (int s = BLK / 2; s > 0; s >>= 1) {
    if (tid < s) {
#pragma unroll
      for (int b = 0; b < NB; ++b) {
        s_conf[b * BLK + tid] += s_conf[b * BLK + tid + s];
        s_cp[b * BLK + tid] += s_cp[b * BLK + tid + s];
      }
    }
    __syncthreads();
  }

  // Deterministic cross-block merge: integer atomics only.
  if (tid < NB) {
    unsigned cp = s_cp[tid * BLK];
    atomicAdd(&ws->cnt[tid], cp >> 16);
    atomicAdd(&ws->pos[tid], cp & 0xFFFFu);
    // block conf sum < 16384, convert exactly in double to 2^-24 fixed point
    unsigned long long fx =
        (unsigned long long)((double)s_conf[tid * BLK] * 16777216.0 + 0.5);
    atomicAdd(&ws->conf_fix[tid], fx);
  }
}

__global__ void rd_final(const RdWs* __restrict__ ws, float* __restrict__ out) {
  int b = threadIdx.x;
  if (b < NB) {
    unsigned c = ws->cnt[b];
    float cnt = (float)c;
    float denom = fmaxf(cnt, 1.0f);
    float pos = (float)ws->pos[b];
    float conf = (float)((double)ws->conf_fix[b] * (1.0 / 16777216.0));
    bool ne = c > 0u;
    out[b] = ne ? pos / denom : 0.0f;       // positives_per_bin
    out[NB + b] = ne ? conf / denom : 0.0f; // confidence_per_bin
  }
}

extern "C" void kernel_launch(void* const* d_in, const int* in_sizes, int n_in,
                              void* d_out, int out_size, void* d_ws, size_t ws_size,
                              hipStream_t stream) {
  const float* logits = (const float*)d_in[0];
  const int* labels = (const int*)d_in[1];
  int n = in_sizes[0];
  RdWs* ws = (RdWs*)d_ws;

  rd_init<<<1, 64, 0, stream>>>(ws);

  int n4 = n >> 2;
  int blocks = (n4 + BLK - 1) / BLK;
  if (blocks > MAXBLOCKS) blocks = MAXBLOCKS;  // 64 elems/thread at n=2^25
  if (blocks < 1) blocks = 1;
  rd_accum<<<blocks, BLK, 0, stream>>>(logits, labels, ws, n);

  rd_final<<<1, 32, 0, stream>>>(ws, (float*)d_out);
}